// PIDNets_Adam_Optimizer_51075751084564
// MI455X (gfx1250) — compile-verified
//
#include <hip/hip_runtime.h>

// ---------------------------------------------------------------------------
// PID (partial information decomposition) solver for MI455X / gfx1250.
// One persistent 1024-thread workgroup (32 wave32 waves).  Wave w owns the
// 32x32 y-slice Q[:,:,w] in registers laid out as V_WMMA_F32_16X16X4_F32
// B-fragments, so the Sinkhorn "sum over x1" marginal is a ones-matrix WMMA
// with zero repacking; "sum over x2" is a ds_swizzle xor butterfly.
// Adam state lives in d_ws (L2-resident: 192 MB L2 >> 512 KB working set).
// v4: all hot-loop addressing strength-reduced to one base register plus
// compile-time immediate offsets (cpart folding) to kill the last spills.
// ---------------------------------------------------------------------------

typedef float v2f __attribute__((ext_vector_type(2)));
typedef float v8f __attribute__((ext_vector_type(8)));

#define EPSF   1e-12f
#define TOLF   1e-5f
#define LRF    0.1f
#define B1F    0.9f
#define B2F    0.999f
#define AEPSF  1e-8f
#define MAXIT  100
#define SINKIT 10
#define N2B    1.44269504088896340736f   // 1/ln(2)

#define IDX_LOG  0
#define IDX_M    32768
#define IDX_V    65536
#define IDX_QOLD 98304
#define IDX_QST  0        // q_star dump aliases logits (dead after final softmax)

// Per-thread fragment walk.  Element (x1,x2,y) with
//   x1 = 4*kc + 2*hi + v,  x2 = 16*nc + ll,  y = w
// matches the WMMA 4x16 f32 B-fragment layout per lane.
// Addressing is split into a loop-invariant base (t12b / baseo, one VGPR)
// plus a compile-time constant cpart, so every LDS/global access folds the
// constant into the instruction's immediate offset field.
#define FOR_FRAG(...)                                                        \
  _Pragma("unroll")                                                          \
  for (int kc = 0; kc < 8; ++kc) {                                           \
    _Pragma("unroll")                                                        \
    for (int nc = 0; nc < 2; ++nc) {                                         \
      _Pragma("unroll")                                                      \
      for (int v = 0; v < 2; ++v) {                                          \
        const int x1 = 4 * kc + 2 * hi + v;                                  \
        const int x2 = 16 * nc + ll;                                         \
        const int cpart = ((4 * kc + v) << 5) | (nc << 4);                   \
        const int idx  = baseo + (cpart << 5);                               \
        const int t12i = t12b + cpart;                                       \
        (void)x1; (void)x2; (void)idx; (void)t12i;                           \
        __VA_ARGS__                                                          \
      } } }

// xor-lane exchange via DS_SWIZZLE group-of-32 mode: no index VGPR needed.
// offset = {1'b0, xor_mask[4:0], or_mask[4:0]=0, and_mask[4:0]=0x1F}
template <int MASK>
__device__ __forceinline__ float swzx(float x) {
  return __int_as_float(
      __builtin_amdgcn_ds_swizzle(__float_as_int(x), (MASK << 10) | 0x1F));
}

__device__ __forceinline__ float wsum(float x) {
  x += swzx<1>(x); x += swzx<2>(x); x += swzx<4>(x);
  x += swzx<8>(x); x += swzx<16>(x);
  return x;
}
__device__ __forceinline__ float wmaxr(float x) {
  x = fmaxf(x, swzx<1>(x));  x = fmaxf(x, swzx<2>(x));
  x = fmaxf(x, swzx<4>(x));  x = fmaxf(x, swzx<8>(x));
  x = fmaxf(x, swzx<16>(x));
  return x;
}
__device__ __forceinline__ float hsum16(float x) {  // sum over 16 lanes of this half
  x += swzx<1>(x); x += swzx<2>(x); x += swzx<4>(x); x += swzx<8>(x);
  return x;
}

__device__ __forceinline__ float bsum(float x, float* red) {
  x = wsum(x);
  const int w = threadIdx.x >> 5, l = threadIdx.x & 31;
  __syncthreads();
  if (l == 0) red[w] = x;
  __syncthreads();
  return wsum(red[l]);
}
__device__ __forceinline__ float bmax(float x, float* red) {
  x = wmaxr(x);
  const int w = threadIdx.x >> 5, l = threadIdx.x & 31;
  __syncthreads();
  if (l == 0) red[w] = x;
  __syncthreads();
  return wmaxr(red[l]);
}

// s0[nc] = sum over x1 of q (column sums), via ones(16x4) x B(4x16) WMMA chain.
// D f32 layout puts column N = lane&15 in every lane -> exactly what the
// per-thread update needs, no shuffles.
__device__ __forceinline__ void colsum_x1(const float q[8][2][2], float s0[2]) {
  v2f ones; ones[0] = 1.0f; ones[1] = 1.0f;
#pragma unroll
  for (int nc = 0; nc < 2; ++nc) {
    v8f acc;
#pragma unroll
    for (int i = 0; i < 8; ++i) acc[i] = 0.0f;
#pragma unroll
    for (int kc = 0; kc < 8; ++kc) {
      v2f b; b[0] = q[kc][nc][0]; b[1] = q[kc][nc][1];
      acc = __builtin_amdgcn_wmma_f32_16x16x4_f32(false, ones, false, b,
                                                  (short)0, acc, false, false);
    }
    s0[nc] = acc[0];
  }
}

// s1[kc][v] = sum over x2 (row sums) -- standalone version for the prologue.
__device__ __forceinline__ void rowsum_x2(const float q[8][2][2], float s1[8][2]) {
#pragma unroll
  for (int kc = 0; kc < 8; ++kc)
#pragma unroll
    for (int v = 0; v < 2; ++v)
      s1[kc][v] = hsum16(q[kc][0][v] + q[kc][1][v]);
}

__device__ __forceinline__ void sinkhorn10(float q[8][2][2], const float px2y[2],
                                           const float px1y[8][2]) {
#pragma unroll 1
  for (int it = 0; it < SINKIT; ++it) {
    float s0[2];
    colsum_x1(q, s0);
    const float f0a = px2y[0] / (s0[0] + EPSF);
    const float f0b = px2y[1] / (s0[1] + EPSF);
#pragma unroll
    for (int kc = 0; kc < 8; ++kc)
#pragma unroll
      for (int v = 0; v < 2; ++v) { q[kc][0][v] *= f0a; q[kc][1][v] *= f0b; }
    // fused row-sum + scale: no s1[] array, one scalar temp per (kc,v)
#pragma unroll
    for (int kc = 0; kc < 8; ++kc)
#pragma unroll
      for (int v = 0; v < 2; ++v) {
        const float s1 = hsum16(q[kc][0][v] + q[kc][1][v]);
        const float f1 = px1y[kc][v] / (s1 + EPSF);
        q[kc][0][v] *= f1; q[kc][1][v] *= f1;
      }
  }
}

// MI (bits) of a 32x32 LDS joint; whole wave participates, lane l = row l.
__device__ __forceinline__ float mi32_lds(const float* M, int l) {
  float ra = 0.f, cb = 0.f;
#pragma unroll 1
  for (int j = 0; j < 32; ++j) { ra += M[(l << 5) + j]; cb += M[(j << 5) + l]; }
  const float tot = wsum(ra) + EPSF;
  const float la = logf(ra / tot + EPSF);
  float acc = 0.f;
#pragma unroll 1
  for (int j = 0; j < 32; ++j) {
    const float pv = M[(l << 5) + j] / tot;
    const float mb = __shfl(cb, j, 32) / tot;
    acc += pv * (logf(pv + EPSF) - la - logf(mb + EPSF));
  }
  return wsum(acc) * N2B;
}

// MI (bits) of a strided 32x32 joint in global memory (L2-resident).
__device__ __forceinline__ float mi32_g(const float* base, int sr, int scn, int l) {
  float ra = 0.f, cb = 0.f;
#pragma unroll 1
  for (int j = 0; j < 32; ++j) { ra += base[l * sr + j * scn]; cb += base[j * sr + l * scn]; }
  const float tot = wsum(ra) + EPSF;
  const float la = logf(ra / tot + EPSF);
  float acc = 0.f;
#pragma unroll 1
  for (int j = 0; j < 32; ++j) {
    const float pv = base[l * sr + j * scn] / tot;
    const float mb = __shfl(cb, j, 32) / tot;
    acc += pv * (logf(pv + EPSF) - la - logf(mb + EPSF));
  }
  return wsum(acc) * N2B;
}

__global__ __launch_bounds__(1024, 1) void pid_solver(const float* __restrict__ gP,
                                                      float* __restrict__ out,
                                                      float* __restrict__ ws) {
  __shared__ float red[32];
  __shared__ float T1[1024];   // [x1][y]
  __shared__ float T2[1024];   // [x2][y]
  __shared__ float T12[1024];  // [x1][x2]
  __shared__ float sc[4];      // mi1, mi2, cmi1, cmi0

  const int tid = threadIdx.x;
  const int w  = tid >> 5;     // wave id == y
  const int l  = tid & 31;
  const int hi = l >> 4;
  const int ll = l & 15;
  const int t12b  = (hi << 6) | ll;        // runtime part of (x1<<5)|x2
  const int baseo = (t12b << 5) | w;       // runtime part of the flat index

  __builtin_prefetch(gP + tid * 32, 0, 1);

  // ---- prologue: P marginals from raw loads, rescaled by 1/(sum+eps).
  //      (identical to normalizing P first; keeps pf's live range short) ----
  float px2y[2];
  float px1y[8][2];
  float py;
  {
    float pf[8][2][2];
    float loc = 0.f;
    FOR_FRAG(pf[kc][nc][v] = gP[idx]; loc += pf[kc][nc][v];)
    const float ptot = bsum(loc, red) + EPSF;
    colsum_x1(pf, px2y);                 // raw P_x2y[x2][y]
    rowsum_x2(pf, px1y);                 // raw P_x1y[x1][y]
    const float inv = 1.f / ptot;
    px2y[0] *= inv; px2y[1] *= inv;
#pragma unroll
    for (int kc = 0; kc < 8; ++kc) { px1y[kc][0] *= inv; px1y[kc][1] *= inv; }
    float t = 0.f;
#pragma unroll
    for (int kc = 0; kc < 8; ++kc) t += px1y[kc][0] + px1y[kc][1];
    py = t + swzx<16>(t);                // add the other half's x1 set
  } // pf dead here

  // ---- warm start Q0, logits/m/v/q_old init ----
  float qf[8][2][2];
  float loc = 0.f;
  FOR_FRAG(const float q0 = px1y[kc][v] * px2y[nc] / (py + EPSF);
           qf[kc][nc][v] = q0; loc += q0;)
  const float q0tot = bsum(loc, red) + EPSF;
  FOR_FRAG(const float q0 = qf[kc][nc][v] / q0tot;
           qf[kc][nc][v] = q0;
           ws[IDX_LOG  + idx] = logf(fmaxf(q0, EPSF));
           ws[IDX_M    + idx] = 0.f;
           ws[IDX_V    + idx] = 0.f;
           ws[IDX_QOLD + idx] = q0;)

  // ---- Adam loop ----
  int   iters = 0;
  bool  done  = false;
  float b1p = 1.f, b2p = 1.f;

#pragma unroll 1
  for (int i = 0; i < MAXIT; ++i) {
    // softmax(logits): probs live only in qf; (mx, se) retained as scalars so
    // the gradient passes can recompute any prob with one load + one exp.
    float mx = -3.0e38f;
    FOR_FRAG(mx = fmaxf(mx, ws[IDX_LOG + idx]);)
    mx = bmax(mx, red);
    loc = 0.f;
    FOR_FRAG(const float e = expf(ws[IDX_LOG + idx] - mx);
             qf[kc][nc][v] = e; loc += e;)
    const float se = bsum(loc, red);
    FOR_FRAG(qf[kc][nc][v] /= se;)

    // sinkhorn projection (barrier-free, per-wave) + global normalize
    sinkhorn10(qf, px2y, px1y);
    loc = 0.f;
    FOR_FRAG(loc += qf[kc][nc][v];)
    const float st = bsum(loc, red) + EPSF;
    FOR_FRAG(qf[kc][nc][v] /= st;)

    // convergence check (matches reference masking semantics)
    float ch = 0.f;
    FOR_FRAG(ch = fmaxf(ch, fabsf(qf[kc][nc][v] - ws[IDX_QOLD + idx]));)
    ch = bmax(ch, red);
    iters += done ? 0 : 1;
    done = done || ((ch < TOLF) && (i > 1));
    if (done) break;                       // uniform across the block

    // Qx[x1][x2] = sum_y q_proj
    T12[tid] = 0.f;
    __syncthreads();
    FOR_FRAG(atomicAdd(&T12[t12i], qf[kc][nc][v]);)
    __syncthreads();

    // approx gradient: dL/dQ* ~ log(Q*/Qx), straight-through sinkhorn scale,
    // exact softmax backward (needs global dot).
    loc = 0.f;
    FOR_FRAG(const float qsv = expf(ws[IDX_LOG + idx] - mx) / se;
             const float Qx = T12[t12i];
             const float r  = qf[kc][nc][v] / (Qx + EPSF);
             const float gq = logf(r + EPSF) * (qf[kc][nc][v] / (qsv + EPSF));
             loc += qsv * gq;)
    const float dot = bsum(loc, red);

    const float b1pn = b1p * B1F, b2pn = b2p * B2F;
    const float ibc1 = 1.f / (1.f - b1pn), ibc2 = 1.f / (1.f - b2pn);
    FOR_FRAG(const float lg  = ws[IDX_LOG + idx];
             const float qsv = expf(lg - mx) / se;
             const float Qx = T12[t12i];
             const float r  = qf[kc][nc][v] / (Qx + EPSF);
             const float gq = logf(r + EPSF) * (qf[kc][nc][v] / (qsv + EPSF));
             const float g  = qsv * (gq - dot);
             const float mn = B1F * ws[IDX_M + idx] + (1.f - B1F) * g;
             const float vn = B2F * ws[IDX_V + idx] + (1.f - B2F) * g * g;
             ws[IDX_M   + idx] = mn;
             ws[IDX_V   + idx] = vn;
             ws[IDX_LOG + idx] = lg - LRF * (mn * ibc1) / (sqrtf(vn * ibc2) + AEPSF);
             ws[IDX_QOLD + idx] = qf[kc][nc][v];)
    b1p = b1pn; b2p = b2pn;
  }

  // ---- final forward with converged logits ----
  {
    float mx = -3.0e38f;
    FOR_FRAG(mx = fmaxf(mx, ws[IDX_LOG + idx]);)
    mx = bmax(mx, red);
    loc = 0.f;
    FOR_FRAG(const float e = expf(ws[IDX_LOG + idx] - mx);
             qf[kc][nc][v] = e; loc += e;)
    const float se = bsum(loc, red);
    FOR_FRAG(qf[kc][nc][v] /= se;)
    sinkhorn10(qf, px2y, px1y);
    loc = 0.f;
    FOR_FRAG(loc += qf[kc][nc][v];)
    const float st = bsum(loc, red) + EPSF;
    FOR_FRAG(qf[kc][nc][v] /= st;)
  }

  // dump q_star (logits region is dead now)
  FOR_FRAG(ws[IDX_QST + idx] = qf[kc][nc][v];)
  __threadfence();
  __syncthreads();

  // ---- metric tables from q_star ----
  T1[tid] = 0.f; T2[tid] = 0.f; T12[tid] = 0.f;
  if (tid < 4) sc[tid] = 0.f;
  __syncthreads();
  FOR_FRAG(const float q = qf[kc][nc][v];
           atomicAdd(&T1[(x1 << 5) | w], q);
           atomicAdd(&T2[(x2 << 5) | w], q);
           atomicAdd(&T12[t12i], q);)
  __syncthreads();

  if (w == 0) { const float r = mi32_lds(T1, l); if (l == 0) sc[0] = r; }
  if (w == 1) { const float r = mi32_lds(T2, l); if (l == 0) sc[1] = r; }

  // CMI cond on x2: wave w handles slice x2 == w, joint [y][x1]
  {
    const float wc  = wsum(T2[(w << 5) | l]);
    const float mis = mi32_g(ws + IDX_QST + (w << 5), 1, 1024, l);
    if (l == 0 && wc > EPSF) atomicAdd(&sc[2], wc * mis);
  }
  // CMI cond on x1: wave w handles slice x1 == w, joint [y][x2]
  {
    const float wc  = wsum(T1[(w << 5) | l]);
    const float mis = mi32_g(ws + IDX_QST + (w << 10), 1, 32, l);
    if (l == 0 && wc > EPSF) atomicAdd(&sc[3], wc * mis);
  }

  // MI of the (y, x1*x2) joint for q_star
  const float may = wsum(T1[(l << 5) | w]);   // ma[y=w] = sum_x1 T1[x1][w]
  float t12tot;
  {
    float t = 0.f;
#pragma unroll 1
    for (int j = 0; j < 32; ++j) t += T12[(l << 5) | j];
    t12tot = wsum(t);
  }
  {
    const float tot = t12tot + EPSF;
    const float lma = logf(may / tot + EPSF);
    loc = 0.f;
    FOR_FRAG(const float pv = qf[kc][nc][v] / tot;
             loc += pv * (logf(pv + EPSF) - lma -
                          logf(T12[t12i] / tot + EPSF));)
  }
  const float mi12q = bsum(loc, red) * N2B;

  // MI of the (y, x1*x2) joint for P: reload raw P from L2 (MI normalizes
  // internally, so the raw tensor is equivalent to the normalized one).
  __syncthreads();
  T12[tid] = 0.f;
  __syncthreads();
  loc = 0.f;
  FOR_FRAG(const float pv = gP[idx];
           atomicAdd(&T12[t12i], pv);
           loc += pv;)
  __syncthreads();
  const float mayp  = wsum(loc);
  const float ptot2 = bsum(loc, red) + EPSF;
  {
    const float lma = logf(mayp / ptot2 + EPSF);
    float acc = 0.f;
    FOR_FRAG(const float pv = gP[idx] / ptot2;
             acc += pv * (logf(pv + EPSF) - lma -
                          logf(T12[t12i] / ptot2 + EPSF));)
    loc = acc;
  }
  const float mi12p = bsum(loc, red) * N2B;

  __syncthreads();
  if (tid == 0) {
    out[0] = sc[0] + sc[1] - mi12q;   // R  (co-information)
    out[1] = sc[2];                   // U1 (cmi cond on x2)
    out[2] = sc[3];                   // U2 (cmi cond on x1)
    out[3] = mi12p - mi12q;           // S  (synergy)
    out[4] = (float)iters;
  }
}

extern "C" void kernel_launch(void* const* d_in, const int* in_sizes, int n_in,
                              void* d_out, int out_size, void* d_ws, size_t ws_size,
                              hipStream_t stream) {
  (void)in_sizes; (void)n_in; (void)out_size; (void)ws_size;
  const float* P = (const float*)d_in[0];
  float* out = (float*)d_out;
  float* ws  = (float*)d_ws;
  pid_solver<<<dim3(1), dim3(1024), 0, stream>>>(P, out, ws);
}